// EncapsulatedMultiHeadAttention_60206851555581
// MI455X (gfx1250) — compile-verified
//
#include <hip/hip_runtime.h>

// ---------------------------------------------------------------------------
// MHA forward for MI455X (gfx1250, wave32) using native fp32 WMMA
// (V_WMMA_F32_16X16X4_F32). Problem is HBM-bound (~26 GFLOP vs ~250MB
// traffic @ 23.3 TB/s), so fp32 matrix ops keep reference precision with
// no throughput penalty.
// ---------------------------------------------------------------------------

typedef __attribute__((ext_vector_type(2))) float v2f;
typedef __attribute__((ext_vector_type(8))) float v8f;

#define N_EMBD   384
#define N_HEAD   6
#define HEAD_DIM 64
#define T_SEQ    256
#define BATCH    64
#define ROWS     (BATCH * T_SEQ)     /* 16384 */
#define QKV_N    (3 * N_EMBD)        /* 1152  */

__device__ __forceinline__ v8f wmma_f32(v2f a, v2f b, v8f c) {
  // 8-arg form: (neg_a, A, neg_b, B, c_mod, C, reuse_a, reuse_b)
  return __builtin_amdgcn_wmma_f32_16x16x4_f32(false, a, false, b,
                                               (short)0, c, false, false);
}

// ---------------------------------------------------------------------------
// Kernel 1: qkv = x @ W_qkv + b_qkv, scattered into [sel][B,H,T,D] layout.
// Wave tile: 16 rows x 64 cols (one full head-dim group => sel/h constant).
// ---------------------------------------------------------------------------
__global__ __launch_bounds__(256) void qkv_gemm_kernel(
    const float* __restrict__ x, const float* __restrict__ W,
    const float* __restrict__ bias, float* __restrict__ qkv) {
  const int lane = threadIdx.x & 31;
  const int wid  = blockIdx.x * 8 + (threadIdx.x >> 5);
  const int NG   = QKV_N / 64;                 // 18 col-groups
  const int mTile = wid / NG;
  const int g     = wid % NG;
  if (mTile >= ROWS / 16) return;
  const int rowBase = mTile * 16;
  const int nBase   = g * 64;
  const int half = lane >> 4;
  const int lm   = lane & 15;

  v8f acc0 = {}, acc1 = {}, acc2 = {}, acc3 = {};
  const float* Arow = x + (size_t)(rowBase + lm) * N_EMBD;

  for (int k = 0; k < N_EMBD; k += 4) {
    __builtin_prefetch(Arow + k + 32, 0, 0);
    const int ko = k + half * 2;
    // A frag: lane holds A[M=lm][ko], A[M=lm][ko+1]
    v2f a = *(const v2f*)(Arow + ko);
    // B frag: lane holds B[ko][n], B[ko+1][n] ; n = nBase + j*16 + lm
    const float* B0 = W + (size_t)ko * QKV_N + nBase + lm;
    const float* B1 = B0 + QKV_N;
    v2f b0 = { B0[0],  B1[0]  };
    v2f b1 = { B0[16], B1[16] };
    v2f b2 = { B0[32], B1[32] };
    v2f b3 = { B0[48], B1[48] };
    acc0 = wmma_f32(a, b0, acc0);
    acc1 = wmma_f32(a, b1, acc1);
    acc2 = wmma_f32(a, b2, acc2);
    acc3 = wmma_f32(a, b3, acc3);
  }

  const float bs0 = bias[nBase + lm];
  const float bs1 = bias[nBase + 16 + lm];
  const float bs2 = bias[nBase + 32 + lm];
  const float bs3 = bias[nBase + 48 + lm];
  const int sel = g / N_HEAD;
  const int h   = g % N_HEAD;
  const size_t QSZ = (size_t)ROWS * N_EMBD;    // one of q/k/v
  float* dstbase = qkv + (size_t)sel * QSZ;

#pragma unroll
  for (int i = 0; i < 8; i++) {
    // C/D layout: lane half selects row+8, vgpr i is row i
    const int r  = rowBase + i + 8 * half;     // global token row
    const int bb = r >> 8;                     // batch
    const int tt = r & 255;                    // seq pos
    float* o = dstbase + (((size_t)(bb * N_HEAD + h)) * T_SEQ + tt) * HEAD_DIM;
    o[lm]      = acc0[i] + bs0;
    o[16 + lm] = acc1[i] + bs1;
    o[32 + lm] = acc2[i] + bs2;
    o[48 + lm] = acc3[i] + bs3;
  }
}

// ---------------------------------------------------------------------------
// Kernel 2: causal flash-attention per (b,h). One wave per 16-row Q block.
// Scores via WMMA (K read directly as B-frag == K^T), online softmax with
// shfl_xor row reductions, P->A-layout exchange through per-wave LDS tile
// ordered with s_wait_dscnt (LDS in-order per wave, no workgroup barrier).
// ---------------------------------------------------------------------------
__global__ __launch_bounds__(256) void attn_kernel(
    const float* __restrict__ qkv, float* __restrict__ yout) {
  __shared__ float pshare[8][16 * 17];         // 17-stride pad kills bank conflicts
  const int lane = threadIdx.x & 31;
  const int w    = threadIdx.x >> 5;
  const int wid  = blockIdx.x * 8 + w;
  const int qb   = wid & 15;                   // q block within sequence
  const int bh   = wid >> 4;                   // fused (batch, head)
  if (bh >= BATCH * N_HEAD) return;
  const size_t QSZ = (size_t)ROWS * N_EMBD;
  const float* q    = qkv + (size_t)bh * T_SEQ * HEAD_DIM;
  const float* kmat = q + QSZ;
  const float* vmat = q + 2 * QSZ;
  const int half = lane >> 4;
  const int lm   = lane & 15;
  float* P = pshare[w];

  v8f o0 = {}, o1 = {}, o2 = {}, o3 = {};
  float mrow[8], lrow[8];
#pragma unroll
  for (int i = 0; i < 8; i++) { mrow[i] = -1e30f; lrow[i] = 0.f; }

  // Preload all Q fragments for this 16-row block (16 x v2f, D=64)
  v2f qf[16];
#pragma unroll
  for (int kk = 0; kk < 16; kk++)
    qf[kk] = *(const v2f*)(q + (size_t)(qb * 16 + lm) * HEAD_DIM + kk * 4 + half * 2);

  const float scale = 0.125f;                  // 1/sqrt(64)
  const int colg_base = lm;

  for (int kb = 0; kb <= qb; kb++) {
    // ---- S = Q @ K^T : B-frag lane holds K[kb*16+lm][d], K[..][d+1]
    v8f s = {};
#pragma unroll
    for (int kk = 0; kk < 16; kk++) {
      v2f bf = *(const v2f*)(kmat + (size_t)(kb * 16 + lm) * HEAD_DIM + kk * 4 + half * 2);
      s = wmma_f32(qf[kk], bf, s);
    }
    // ---- scale + causal mask
    const int colg = kb * 16 + colg_base;
#pragma unroll
    for (int i = 0; i < 8; i++) {
      const int rowg = qb * 16 + i + 8 * half;
      float v = s[i] * scale;
      if (colg > rowg) v = -1e30f;
      s[i] = v;
    }
    // ---- online softmax: row max / correction / exp / row sum
    float corr[8];
#pragma unroll
    for (int i = 0; i < 8; i++) {
      float v = s[i];
#pragma unroll
      for (int off = 1; off < 16; off <<= 1)
        v = fmaxf(v, __shfl_xor(v, off, 32));  // within 16-lane half
      const float mnew = fmaxf(mrow[i], v);
      corr[i] = __expf(mrow[i] - mnew);
      mrow[i] = mnew;
    }
#pragma unroll
    for (int i = 0; i < 8; i++) {
      const float p = __expf(s[i] - mrow[i]);
      s[i] = p;
      float rs = p;
#pragma unroll
      for (int off = 1; off < 16; off <<= 1)
        rs += __shfl_xor(rs, off, 32);
      lrow[i] = lrow[i] * corr[i] + rs;
      o0[i] *= corr[i]; o1[i] *= corr[i]; o2[i] *= corr[i]; o3[i] *= corr[i];
    }
    // ---- C/D layout -> A layout exchange through per-wave LDS tile
#pragma unroll
    for (int i = 0; i < 8; i++)
      P[(i + 8 * half) * 17 + lm] = s[i];
    __builtin_amdgcn_wave_barrier();
    asm volatile("s_wait_dscnt 0x0" ::: "memory");
    // ---- O += P @ V : A-frag from LDS, V read directly as B-frag
#pragma unroll
    for (int kk = 0; kk < 4; kk++) {
      v2f af;
      af.x = P[lm * 17 + kk * 4 + half * 2];
      af.y = P[lm * 17 + kk * 4 + half * 2 + 1];
      const float* V0 = vmat + (size_t)(kb * 16 + kk * 4 + half * 2) * HEAD_DIM + lm;
      const float* V1 = V0 + HEAD_DIM;
      v2f b0 = { V0[0],  V1[0]  };
      v2f b1 = { V0[16], V1[16] };
      v2f b2 = { V0[32], V1[32] };
      v2f b3 = { V0[48], V1[48] };
      o0 = wmma_f32(af, b0, o0);
      o1 = wmma_f32(af, b1, o1);
      o2 = wmma_f32(af, b2, o2);
      o3 = wmma_f32(af, b3, o3);
    }
    __builtin_amdgcn_wave_barrier();
    asm volatile("s_wait_dscnt 0x0" ::: "memory"); // WAR: next store vs these loads
  }

  // ---- normalize + write back to [B*T, C] with head offset
  const int bb = bh / N_HEAD;
  const int hh = bh % N_HEAD;
#pragma unroll
  for (int i = 0; i < 8; i++) {
    const float inv = 1.0f / lrow[i];
    const int rowg = qb * 16 + i + 8 * half;   // t
    float* o = yout + ((size_t)(bb * T_SEQ) + rowg) * N_EMBD + hh * HEAD_DIM;
    o[lm]      = o0[i] * inv;
    o[16 + lm] = o1[i] * inv;
    o[32 + lm] = o2[i] * inv;
    o[48 + lm] = o3[i] * inv;
  }
}

// ---------------------------------------------------------------------------
// Kernel 3: out = attn @ W_proj + b_proj  (16x64 wave tiles, fp32 WMMA)
// ---------------------------------------------------------------------------
__global__ __launch_bounds__(256) void proj_gemm_kernel(
    const float* __restrict__ a_in, const float* __restrict__ W,
    const float* __restrict__ bias, float* __restrict__ out) {
  const int lane = threadIdx.x & 31;
  const int wid  = blockIdx.x * 8 + (threadIdx.x >> 5);
  const int NG   = N_EMBD / 64;                // 6
  const int mTile = wid / NG;
  const int g     = wid % NG;
  if (mTile >= ROWS / 16) return;
  const int rowBase = mTile * 16;
  const int nBase   = g * 64;
  const int half = lane >> 4;
  const int lm   = lane & 15;

  v8f acc0 = {}, acc1 = {}, acc2 = {}, acc3 = {};
  const float* Arow = a_in + (size_t)(rowBase + lm) * N_EMBD;

  for (int k = 0; k < N_EMBD; k += 4) {
    __builtin_prefetch(Arow + k + 32, 0, 0);
    const int ko = k + half * 2;
    v2f a = *(const v2f*)(Arow + ko);
    const float* B0 = W + (size_t)ko * N_EMBD + nBase + lm;
    const float* B1 = B0 + N_EMBD;
    v2f b0 = { B0[0],  B1[0]  };
    v2f b1 = { B0[16], B1[16] };
    v2f b2 = { B0[32], B1[32] };
    v2f b3 = { B0[48], B1[48] };
    acc0 = wmma_f32(a, b0, acc0);
    acc1 = wmma_f32(a, b1, acc1);
    acc2 = wmma_f32(a, b2, acc2);
    acc3 = wmma_f32(a, b3, acc3);
  }

  const float bs0 = bias[nBase + lm];
  const float bs1 = bias[nBase + 16 + lm];
  const float bs2 = bias[nBase + 32 + lm];
  const float bs3 = bias[nBase + 48 + lm];
#pragma unroll
  for (int i = 0; i < 8; i++) {
    const int r = rowBase + i + 8 * half;
    float* o = out + (size_t)r * N_EMBD + nBase;
    o[lm]      = acc0[i] + bs0;
    o[16 + lm] = acc1[i] + bs1;
    o[32 + lm] = acc2[i] + bs2;
    o[48 + lm] = acc3[i] + bs3;
  }
}

// ---------------------------------------------------------------------------
extern "C" void kernel_launch(void* const* d_in, const int* in_sizes, int n_in,
                              void* d_out, int out_size, void* d_ws, size_t ws_size,
                              hipStream_t stream) {
  const float* x      = (const float*)d_in[0];
  const float* W_qkv  = (const float*)d_in[1];
  const float* b_qkv  = (const float*)d_in[2];
  const float* W_proj = (const float*)d_in[3];
  const float* b_proj = (const float*)d_in[4];
  float* out = (float*)d_out;

  const size_t QSZ = (size_t)ROWS * N_EMBD;    // per q/k/v buffer (floats)
  float* qkv  = (float*)d_ws;                  // 3 * QSZ floats
  float* attn = qkv + 3 * QSZ;                 // QSZ floats

  const dim3 blk(256);
  const int waves1 = (ROWS / 16) * (QKV_N / 64);   // 18432
  qkv_gemm_kernel<<<dim3(waves1 / 8), blk, 0, stream>>>(x, W_qkv, b_qkv, qkv);

  const int waves2 = BATCH * N_HEAD * (T_SEQ / 16); // 6144
  attn_kernel<<<dim3(waves2 / 8), blk, 0, stream>>>(qkv, attn);

  const int waves3 = (ROWS / 16) * (N_EMBD / 64);   // 6144
  proj_gemm_kernel<<<dim3(waves3 / 8), blk, 0, stream>>>(attn, W_proj, b_proj, out);
}